// GNN_47528108097587
// MI455X (gfx1250) — compile-verified
//
#include <hip/hip_runtime.h>
#include <hip/hip_bf16.h>

#define N_NODES 100000
#define N_EDGES 3200000
#define N_GRAPHS 1000
#define D 128
#define OUTF 64
#define NLAYERS 3
#define BN_EPS 1e-5f
#define GPAD 1008  // N_GRAPHS padded to multiple of 16 for WMMA head

typedef float v2f __attribute__((ext_vector_type(2)));
typedef float v8f __attribute__((ext_vector_type(8)));

// ---------------- elementwise helpers ----------------

__global__ void k_zero(float* __restrict__ p, int n) {
  int i = blockIdx.x * blockDim.x + threadIdx.x;
  if (i < n) p[i] = 0.f;
}

__global__ void k_copy(const float4* __restrict__ src, float4* __restrict__ dst, int n4) {
  int i = blockIdx.x * blockDim.x + threadIdx.x;
  if (i < n4) dst[i] = src[i];
}

// ---------------- edge scatter-add: agg[dst] += h[src] ----------------
// one wave per edge; edge id is wave-uniform -> readfirstlane makes the
// src/dst index loads scalar (s_load) and the row gather an SGPR-base load.
// Each lane handles 4 consecutive features (b128 gather, 4 native f32 atomics).
// Working set (51.2MB node matrix) is L2-resident on the 192MB L2.

__global__ __launch_bounds__(256) void k_scatter(const float* __restrict__ h,
                                                 float* __restrict__ agg,
                                                 const int* __restrict__ srcI,
                                                 const int* __restrict__ dstI) {
  int e = __builtin_amdgcn_readfirstlane(blockIdx.x * 8 + (threadIdx.x >> 5));
  if (e >= N_EDGES) return;
  int lane = threadIdx.x & 31;
  int s = srcI[e];
  int d = dstI[e];
  float4 v = *(const float4*)(h + (size_t)s * D + lane * 4);
  float* p = agg + (size_t)d * D + lane * 4;
  unsafeAtomicAdd(p + 0, v.x);
  unsafeAtomicAdd(p + 1, v.y);
  unsafeAtomicAdd(p + 2, v.z);
  unsafeAtomicAdd(p + 3, v.w);
}

// ---------------- fp32 WMMA GEMM: Out = act(A[nrows x 128] @ W[128 x 128] + b) ----
// 8 waves/block, each wave = 16 rows x 128 cols (8 accumulators v8f).
// A fragment (16x4 f32): lanes 0-15 hold K=k,k+1; lanes 16-31 hold K=k+2,k+3
//   -> one contiguous 8B global load per lane.
// B fragment (4x16 f32): W is staged in LDS pre-paired as v2f
//   Wp[k2*128+col] = {W[2*k2][col], W[2*k2+1][col]} so each fragment is one
//   aligned ds_load_b64 directly into the WMMA operand register pair.
// Safe in-place (A == Out): each wave consumes all K of its rows before storing.

__global__ __launch_bounds__(256) void k_gemm128(const float* __restrict__ A,
                                                 const float* __restrict__ W,
                                                 const float* __restrict__ bias,
                                                 float* __restrict__ Out,
                                                 int nrows, int do_relu) {
  __shared__ v2f Wp[(D / 2) * D];  // 64 KB, K-pair-packed
  for (int i = threadIdx.x; i < (D / 2) * D; i += 256) {
    int k2 = i >> 7;    // K-pair index 0..63
    int col = i & 127;  // output column
    v2f t;
    t.x = W[(2 * k2) * D + col];
    t.y = W[(2 * k2 + 1) * D + col];
    Wp[i] = t;
  }
  __syncthreads();

  const int wave = threadIdx.x >> 5;
  const int lane = threadIdx.x & 31;
  const int lane16 = lane & 15;
  const int hi = lane >> 4;  // 0: lanes 0-15 (K=k,k+1), 1: lanes 16-31 (K=k+2,k+3)

  const int row0 = blockIdx.x * 128 + wave * 16;
  int arow = row0 + lane16;
  if (arow > nrows - 1) arow = nrows - 1;  // clamp; clamped results never stored
  const float* Ap = A + (size_t)arow * D + 2 * hi;

  v8f acc[8];
#pragma unroll
  for (int n = 0; n < 8; n++) acc[n] = (v8f){0.f, 0.f, 0.f, 0.f, 0.f, 0.f, 0.f, 0.f};

  for (int k = 0; k < D; k += 4) {
    v2f a = *(const v2f*)(Ap + k);  // 8B aligned: arow*128 + 2*hi + k
    const v2f* brow = &Wp[((k >> 1) + hi) * D + lane16];
#pragma unroll
    for (int n = 0; n < 8; n++) {
      v2f b = brow[n * 16];  // single ds_load_b64 per fragment
      acc[n] = __builtin_amdgcn_wmma_f32_16x16x4_f32(false, a, false, b, (short)0,
                                                     acc[n], false, false);
    }
  }

#pragma unroll
  for (int n = 0; n < 8; n++) {
    const int col = n * 16 + lane16;
    const float bv = bias[col];
#pragma unroll
    for (int v = 0; v < 8; v++) {
      int r = row0 + v + 8 * hi;  // C/D layout: vgpr v -> M=v (lo half) / v+8 (hi half)
      if (r < nrows) {
        float o = acc[n][v] + bv;
        if (do_relu) o = fmaxf(o, 0.f);
        Out[(size_t)r * D + col] = o;
      }
    }
  }
}

// ---------------- BatchNorm: per-feature stats over all nodes ----------------

__global__ void k_bn_stats(const float* __restrict__ h, float* __restrict__ sums,
                           float* __restrict__ sumsq) {
  int f = threadIdx.x;  // 128 threads, coalesced 512B row reads
  float s = 0.f, q = 0.f;
  for (int r = blockIdx.x; r < N_NODES; r += gridDim.x) {
    float v = h[(size_t)r * D + f];
    s += v;
    q += v * v;
  }
  unsafeAtomicAdd(&sums[f], s);
  unsafeAtomicAdd(&sumsq[f], q);
}

__global__ void k_bn_scale(const float* __restrict__ sums, const float* __restrict__ sumsq,
                           const float* __restrict__ gamma, const float* __restrict__ beta,
                           float* __restrict__ scale, float* __restrict__ shift) {
  int f = threadIdx.x;
  const float inv_n = 1.0f / (float)N_NODES;
  float mu = sums[f] * inv_n;
  float var = sumsq[f] * inv_n - mu * mu;  // biased variance (matches jnp.var)
  float sc = gamma[f] * rsqrtf(var + BN_EPS);
  scale[f] = sc;
  shift[f] = beta[f] - mu * sc;
}

__global__ void k_bn_apply(float* __restrict__ h, const float* __restrict__ scale,
                           const float* __restrict__ shift) {
  int idx = blockIdx.x * blockDim.x + threadIdx.x;  // over N_NODES*32
  int n = idx >> 5;
  if (n >= N_NODES) return;
  int q = (idx & 31) * 4;
  float4 v = *(float4*)(h + (size_t)n * D + q);
  v.x = fmaxf(v.x * scale[q + 0] + shift[q + 0], 0.f);
  v.y = fmaxf(v.y * scale[q + 1] + shift[q + 1], 0.f);
  v.z = fmaxf(v.z * scale[q + 2] + shift[q + 2], 0.f);
  v.w = fmaxf(v.w * scale[q + 3] + shift[q + 3], 0.f);
  *(float4*)(h + (size_t)n * D + q) = v;
}

// ---------------- per-graph mean pool (sum + count via atomics) ----------------
// one wave per node; node id and batch[] lookup are wave-uniform -> scalar loads.

__global__ __launch_bounds__(256) void k_pool(const float* __restrict__ h,
                                              const int* __restrict__ batch,
                                              float* __restrict__ pooled,
                                              float* __restrict__ cnts) {
  int n = __builtin_amdgcn_readfirstlane(blockIdx.x * 8 + (threadIdx.x >> 5));
  if (n >= N_NODES) return;
  int lane = threadIdx.x & 31;
  int g = batch[n];
  float4 v = *(const float4*)(h + (size_t)n * D + lane * 4);
  float* p = pooled + (size_t)g * D + lane * 4;
  unsafeAtomicAdd(p + 0, v.x);
  unsafeAtomicAdd(p + 1, v.y);
  unsafeAtomicAdd(p + 2, v.z);
  unsafeAtomicAdd(p + 3, v.w);
  if (lane == 0) unsafeAtomicAdd(&cnts[g], 1.0f);
}

// ---------------- head GEMM: out = (pooled/cnt)[1000x128] @ lin_w[128x64] + lin_b ----

__global__ __launch_bounds__(256) void k_head(const float* __restrict__ pooled,
                                              const float* __restrict__ cnts,
                                              const float* __restrict__ Wlin,
                                              const float* __restrict__ blin,
                                              float* __restrict__ out) {
  const int wave = threadIdx.x >> 5;
  const int lane = threadIdx.x & 31;
  const int lane16 = lane & 15;
  const int hi = lane >> 4;
  const int tile = blockIdx.x * 8 + wave;
  const int row0 = tile * 16;
  if (row0 >= GPAD) return;

  const int arow = row0 + lane16;  // < GPAD, always in-bounds (padded, zeroed)
  const float invc = 1.0f / fmaxf(cnts[arow], 1.0f);  // fold mean into A load
  const float* Ap = pooled + (size_t)arow * D + 2 * hi;

  v8f acc[4];
#pragma unroll
  for (int n = 0; n < 4; n++) acc[n] = (v8f){0.f, 0.f, 0.f, 0.f, 0.f, 0.f, 0.f, 0.f};

  for (int k = 0; k < D; k += 4) {
    v2f a = *(const v2f*)(Ap + k);
    a.x *= invc;
    a.y *= invc;
#pragma unroll
    for (int n = 0; n < 4; n++) {
      const int col = n * 16 + lane16;
      v2f b;
      b.x = Wlin[(k + 2 * hi) * OUTF + col];
      b.y = Wlin[(k + 2 * hi + 1) * OUTF + col];
      acc[n] = __builtin_amdgcn_wmma_f32_16x16x4_f32(false, a, false, b, (short)0,
                                                     acc[n], false, false);
    }
  }

#pragma unroll
  for (int n = 0; n < 4; n++) {
    const int col = n * 16 + lane16;
    const float bv = blin[col];
#pragma unroll
    for (int v = 0; v < 8; v++) {
      int r = row0 + v + 8 * hi;
      if (r < N_GRAPHS) out[(size_t)r * OUTF + col] = acc[n][v] + bv;
    }
  }
}

// ---------------- orchestration ----------------

extern "C" void kernel_launch(void* const* d_in, const int* in_sizes, int n_in,
                              void* d_out, int out_size, void* d_ws, size_t ws_size,
                              hipStream_t stream) {
  const float* x = (const float*)d_in[0];
  const int* ei = (const int*)d_in[1];
  const int* srcI = ei;
  const int* dstI = ei + N_EDGES;
  const int* batch = (const int*)d_in[2];
  const float* W1 = (const float*)d_in[3];
  const float* b1 = (const float*)d_in[4];
  const float* W2 = (const float*)d_in[5];
  const float* b2 = (const float*)d_in[6];
  const float* gamma = (const float*)d_in[7];
  const float* beta = (const float*)d_in[8];
  const float* linw = (const float*)d_in[9];
  const float* linb = (const float*)d_in[10];
  float* out = (float*)d_out;

  // workspace layout (floats): two 51.2MB node buffers + stats + pooled(+pad) + counts
  float* ws = (float*)d_ws;
  float* bufA = ws;
  float* bufB = bufA + (size_t)N_NODES * D;
  float* sums = bufB + (size_t)N_NODES * D;
  float* sumsq = sums + D;
  float* scale = sumsq + D;
  float* shift = scale + D;
  float* pooled = shift + D;
  float* cnts = pooled + (size_t)GPAD * D;

  const dim3 blk(256);
  const int n4 = (N_NODES * D) / 4;
  const int copyGrid = (n4 + 255) / 256;
  const int gemmGrid = (N_NODES + 127) / 128;
  const int ewGrid = (N_NODES * 32 + 255) / 256;
  const int nodeWaveGrid = (N_NODES + 7) / 8;  // one wave per node

  const float* h = x;
  float* bufs[2] = {bufA, bufB};
  for (int l = 0; l < NLAYERS; l++) {
    float* w = bufs[l & 1];
    // agg_plus_self = h + segment_sum(h[src] -> dst)
    k_copy<<<copyGrid, blk, 0, stream>>>((const float4*)h, (float4*)w, n4);
    k_scatter<<<(N_EDGES + 7) / 8, blk, 0, stream>>>(h, w, srcI, dstI);
    // MLP (in-place)
    k_gemm128<<<gemmGrid, blk, 0, stream>>>(w, W1 + l * D * D, b1 + l * D, w, N_NODES, 1);
    k_gemm128<<<gemmGrid, blk, 0, stream>>>(w, W2 + l * D * D, b2 + l * D, w, N_NODES, 0);
    // BatchNorm (training stats) + ReLU, in-place
    k_zero<<<1, 256, 0, stream>>>(sums, 2 * D);
    k_bn_stats<<<512, D, 0, stream>>>(w, sums, sumsq);
    k_bn_scale<<<1, D, 0, stream>>>(sums, sumsq, gamma + l * D, beta + l * D, scale, shift);
    k_bn_apply<<<ewGrid, blk, 0, stream>>>(w, scale, shift);
    h = w;
  }

  // global mean pool + linear head
  k_zero<<<(GPAD * D + GPAD + 255) / 256, blk, 0, stream>>>(pooled, GPAD * D + GPAD);
  k_pool<<<nodeWaveGrid, blk, 0, stream>>>(h, batch, pooled, cnts);
  k_head<<<8, blk, 0, stream>>>(pooled, cnts, linw, linb, out);
}